// ComplexNetNative_89739046683232
// MI455X (gfx1250) — compile-verified
//
#include <hip/hip_runtime.h>

typedef __attribute__((ext_vector_type(2))) float v2f;
typedef __attribute__((ext_vector_type(8))) float v8f;

#define N_CLASSES 2
#define N_FEATURES 100
#define DIM 10
#define KSTEPS 25              // 100 features / K=4 per WMMA
#define BVALS (KSTEPS * 64)    // pre-swizzled B stream: 25 steps x 32 lanes x 2 floats
#define WS_BOFF 256            // ReM occupies ws[0..199]; B stream starts at ws[256]

// ---------------------------------------------------------------------------
// Stage 1: ReM[k][a] = sum_{i,j} Re( conj(psi_i) * A[k,i,j,a] * psi_j )
//        = sum_{i,j} A_re*cr - A_im*ci,  cr+i*ci = conj(psi_i)*psi_j
// ---------------------------------------------------------------------------
__global__ void stage1_bilinear(const float* __restrict__ A_re,
                                const float* __restrict__ A_im,
                                const float* __restrict__ psi_re,
                                const float* __restrict__ psi_im,
                                float* __restrict__ ws_m) {
  int t = blockIdx.x * blockDim.x + threadIdx.x;
  if (t >= N_CLASSES * N_FEATURES) return;
  int k = t / N_FEATURES;
  int a = t % N_FEATURES;
  float pr[DIM], pi[DIM];
#pragma unroll
  for (int i = 0; i < DIM; ++i) { pr[i] = psi_re[i]; pi[i] = psi_im[i]; }
  float acc = 0.f;
  for (int i = 0; i < DIM; ++i) {
    for (int j = 0; j < DIM; ++j) {
      float cr = pr[i] * pr[j] + pi[i] * pi[j];
      float ci = pr[i] * pi[j] - pi[i] * pr[j];
      int idx = ((k * DIM + i) * DIM + j) * N_FEATURES + a;
      acc = fmaf(A_re[idx], cr, acc);
      acc = fmaf(-A_im[idx], ci, acc);
    }
  }
  ws_m[t] = acc;
}

// ---------------------------------------------------------------------------
// Stage 2: swizzle ReM into the wave32 B-operand layout of v_wmma_f32_16x16x4:
//   lane n in [0,15]:  VGPR0 = B[K=0][n], VGPR1 = B[K=1][n]
//   lane 16+n:         VGPR0 = B[K=2][n], VGPR1 = B[K=3][n]
//   B[kk][n] = ReM[n][step*4+kk] for n<2 else 0
// Stored as ws_b[step*64 + lane*2 + comp] so stage 3 reads one float2/lane.
// ---------------------------------------------------------------------------
__global__ void stage2_swizzleB(const float* __restrict__ ws_m,
                                float* __restrict__ ws_b) {
  for (int idx = threadIdx.x; idx < BVALS; idx += blockDim.x) {
    int step   = idx >> 6;
    int within = idx & 63;
    int lane   = within >> 1;
    int comp   = within & 1;
    int n      = lane & 15;
    int khalf  = lane >> 4;
    int a      = step * 4 + khalf * 2 + comp;
    float v = 0.f;
    if (n < N_CLASSES && a < N_FEATURES) v = ws_m[n * N_FEATURES + a];
    ws_b[idx] = v;
  }
}

// ---------------------------------------------------------------------------
// Stage 3: out(T x 2) = x(T x 100) @ ReM^T via V_WMMA_F32_16X16X4_F32.
// 1 wave = one 16-row tile of x; 8 waves / 256-thread block (wave32).
// A-operand per lane: lanes 0-15 load x[row][k0+0..1], lanes 16-31 load
// x[row][k0+2..3] (matches the 32-bit 16x4 A layout). B comes from LDS.
// ---------------------------------------------------------------------------
__global__ __launch_bounds__(256) void stage3_gemm(const float* __restrict__ x,
                                                   const float* __restrict__ ws_b,
                                                   float* __restrict__ out) {
  __shared__ float sB[BVALS];
  for (int i = threadIdx.x; i < BVALS; i += 256) sB[i] = ws_b[i];
  __syncthreads();

  const int wave  = threadIdx.x >> 5;
  const int lane  = threadIdx.x & 31;
  const int tile  = blockIdx.x * 8 + wave;
  const long base = (long)tile * 16;
  const int mrow  = lane & 15;
  const int khalf = lane >> 4;

  const float* xrow = x + (base + mrow) * (long)N_FEATURES + khalf * 2;

  v8f c = {0.f, 0.f, 0.f, 0.f, 0.f, 0.f, 0.f, 0.f};
#pragma unroll
  for (int step = 0; step < KSTEPS; ++step) {
    v2f a = *(const v2f*)(xrow + step * 4);                 // global_load_b64
    v2f b = *(const v2f*)(&sB[step * 64 + lane * 2]);       // ds_load_b64
    // 8 args: (neg_a, A, neg_b, B, c_mod, C, reuse_a, reuse_b)
    c = __builtin_amdgcn_wmma_f32_16x16x4_f32(false, a, false, b,
                                              (short)0, c, false, false);
  }

  // C/D layout: VGPR r -> lanes 0-15: (M=r, N=lane); lanes 16-31: (M=r+8, N=lane-16)
  if (mrow < N_CLASSES) {
    const long r0 = base + khalf * 8;
#pragma unroll
    for (int r = 0; r < 8; ++r) {
      out[(r0 + r) * 2 + mrow] = c[r];
    }
  }
}

// ---------------------------------------------------------------------------
extern "C" void kernel_launch(void* const* d_in, const int* in_sizes, int n_in,
                              void* d_out, int out_size, void* d_ws, size_t ws_size,
                              hipStream_t stream) {
  const float* x      = (const float*)d_in[0];
  const float* A_re   = (const float*)d_in[1];
  const float* A_im   = (const float*)d_in[2];
  const float* psi_re = (const float*)d_in[3];
  const float* psi_im = (const float*)d_in[4];
  float* out = (float*)d_out;
  float* ws  = (float*)d_ws;

  float* ws_m = ws;            // 200 floats (ReM)
  float* ws_b = ws + WS_BOFF;  // 1600 floats (swizzled B stream)

  const int T     = in_sizes[0] / N_FEATURES;  // 262144
  const int tiles = T / 16;                    // 16384

  hipLaunchKernelGGL(stage1_bilinear, dim3(1), dim3(256), 0, stream,
                     A_re, A_im, psi_re, psi_im, ws_m);
  hipLaunchKernelGGL(stage2_swizzleB, dim3(1), dim3(64), 0, stream, ws_m, ws_b);
  hipLaunchKernelGGL(stage3_gemm, dim3(tiles / 8), dim3(256), 0, stream,
                     x, ws_b, out);
}